// GraphTrajSimEncoder_67362267070833
// MI455X (gfx1250) — compile-verified
//
#include <hip/hip_runtime.h>

typedef float v2f __attribute__((ext_vector_type(2)));
typedef float v8f __attribute__((ext_vector_type(8)));

#define NNODES 50000
#define NEDGES 800000
#define FIN    256
#define FEMB   256
#define PEDIM  98
#define KRAW   354            // FIN + PEDIM
#define KCAT   384            // KRAW padded to multiple of 16
#define KPAIR  512            // a1 || a2 packed contraction dim

// ---------------------------------------------------------------------------
// small elementwise / setup kernels
// ---------------------------------------------------------------------------
__global__ void k_fill(float* __restrict__ p, int n, float v) {
  int i = blockIdx.x * blockDim.x + threadIdx.x;
  if (i < n) p[i] = v;
}

__global__ void k_degcount(const int* __restrict__ col, float* __restrict__ deg) {
  int e = blockIdx.x * blockDim.x + threadIdx.x;
  if (e < NEDGES)
    __hip_atomic_fetch_add(&deg[col[e]], 1.0f, __ATOMIC_RELAXED, __HIP_MEMORY_SCOPE_AGENT);
}

__global__ void k_rsqrt_inplace(float* __restrict__ d, int n) {
  int i = blockIdx.x * blockDim.x + threadIdx.x;
  if (i < n) { float v = d[i]; d[i] = (v > 0.f) ? rsqrtf(v) : 0.f; }
}

// xcat[i, 0:256]=x, [256:354]=d2an, [354:384]=0
__global__ void k_xcat(const float* __restrict__ x, const float* __restrict__ d2an,
                       float* __restrict__ xc) {
  long long idx = (long long)blockIdx.x * blockDim.x + threadIdx.x;
  if (idx >= (long long)NNODES * KCAT) return;
  int i = (int)(idx / KCAT), k = (int)(idx % KCAT);
  float v = 0.f;
  if (k < FIN)       v = x[(size_t)i * FIN + k];
  else if (k < KRAW) v = d2an[(size_t)i * PEDIM + (k - FIN)];
  xc[idx] = v;
}

// pad Wnode [256,354] -> [256,384] zero-filled
__global__ void k_padw(const float* __restrict__ w, float* __restrict__ o) {
  int idx = blockIdx.x * blockDim.x + threadIdx.x;
  if (idx >= 256 * KCAT) return;
  int r = idx / KCAT, k = idx % KCAT;
  o[idx] = (k < KRAW) ? w[r * KRAW + k] : 0.f;
}

// pack [W1 | W2] -> [256, 512]
__global__ void k_pair(const float* __restrict__ w1, const float* __restrict__ w2,
                       float* __restrict__ o) {
  int idx = blockIdx.x * blockDim.x + threadIdx.x;
  if (idx >= 256 * KPAIR) return;
  int r = idx / KPAIR, k = idx % KPAIR;
  o[idx] = (k < FIN) ? w1[r * FIN + k] : w2[r * FIN + (k - FIN)];
}

// self-loop init of aggregation buffer A[N,512]:
//   A[i, f]     = dinv[i]^2 * h[i,f]   (deg_norm part, self edge)
//   A[i, 256+f] = h[i,f]               (edge_norm part, attr=1 -> 1.0)
__global__ void k_inita(const float* __restrict__ h, const float* __restrict__ dinv,
                        float* __restrict__ a) {
  long long idx = (long long)blockIdx.x * blockDim.x + threadIdx.x;
  if (idx >= (long long)NNODES * FEMB) return;
  int i = (int)(idx / FEMB), f = (int)(idx % FEMB);
  float hv = h[idx];
  float di = dinv[i];
  a[(size_t)i * KPAIR + f]       = di * di * hv;
  a[(size_t)i * KPAIR + FIN + f] = hv;
}

// edge scatter: 64 threads per edge, 4 features each; fp32 atomics hit L2
__global__ void k_scatter(const float* __restrict__ h,
                          const int* __restrict__ row, const int* __restrict__ col,
                          const float* __restrict__ attr,
                          const float* __restrict__ dinv,
                          float* __restrict__ a) {
  long long idx = (long long)blockIdx.x * blockDim.x + threadIdx.x;
  long long e = idx >> 6;
  if (e >= NEDGES) return;
  int f = (int)(idx & 63) * 4;
  int r = row[e], c = col[e];
  float dn = dinv[r] * dinv[c];
  float w = attr[e];
  float en = (w > 0.f) ? fminf(rsqrtf(w), 1.0f) : 0.f;
  float4 v = *(const float4*)&h[(size_t)r * FEMB + f];
  float* p1 = &a[(size_t)c * KPAIR + f];
  float* p2 = p1 + FIN;
  __hip_atomic_fetch_add(&p1[0], dn * v.x, __ATOMIC_RELAXED, __HIP_MEMORY_SCOPE_AGENT);
  __hip_atomic_fetch_add(&p1[1], dn * v.y, __ATOMIC_RELAXED, __HIP_MEMORY_SCOPE_AGENT);
  __hip_atomic_fetch_add(&p1[2], dn * v.z, __ATOMIC_RELAXED, __HIP_MEMORY_SCOPE_AGENT);
  __hip_atomic_fetch_add(&p1[3], dn * v.w, __ATOMIC_RELAXED, __HIP_MEMORY_SCOPE_AGENT);
  __hip_atomic_fetch_add(&p2[0], en * v.x, __ATOMIC_RELAXED, __HIP_MEMORY_SCOPE_AGENT);
  __hip_atomic_fetch_add(&p2[1], en * v.y, __ATOMIC_RELAXED, __HIP_MEMORY_SCOPE_AGENT);
  __hip_atomic_fetch_add(&p2[2], en * v.z, __ATOMIC_RELAXED, __HIP_MEMORY_SCOPE_AGENT);
  __hip_atomic_fetch_add(&p2[3], en * v.w, __ATOMIC_RELAXED, __HIP_MEMORY_SCOPE_AGENT);
}

// out = 0.5*relu(o0) + 0.5*relu(o1)
__global__ void k_combine(const float* __restrict__ o0, const float* __restrict__ o1,
                          float* __restrict__ out) {
  long long idx = (long long)blockIdx.x * blockDim.x + threadIdx.x;
  if (idx >= (long long)NNODES * FEMB) return;
  out[idx] = 0.5f * fmaxf(o0[idx], 0.f) + 0.5f * fmaxf(o1[idx], 0.f);
}

// ---------------------------------------------------------------------------
// WMMA f32 GEMM:  C[M x 256] = A[M x K] * B[256 x K]^T    (K % 16 == 0)
// block: 256 threads = 8 waves, 64x64 C tile; wave = 16(M) x 32(N)
// uses V_WMMA_F32_16X16X4_F32 (full fp32 matrix pipe, matches reference math)
// ---------------------------------------------------------------------------
__global__ __launch_bounds__(256) void wmma_gemm_nt(
    const float* __restrict__ A, int lda,
    const float* __restrict__ B,   // [256, K], row-major, ldb == K
    float* __restrict__ C, int ldc,
    int M, int K) {
  __shared__ float As[64][20];     // stride 20 floats: pad for LDS banks, keeps 8B align
  __shared__ float Bs[64][20];

  const int tid  = threadIdx.x;
  const int lane = tid & 31;
  const int wave = tid >> 5;
  const int wm   = wave & 3;       // M subtile 0..3
  const int wn   = wave >> 2;      // N subtile 0..1 (32 wide each)
  const int mBase = blockIdx.x * 64;
  const int nBase = blockIdx.y * 64;

  v8f acc0 = {};                   // 16x16 tile at n = wn*32
  v8f acc1 = {};                   // 16x16 tile at n = wn*32 + 16

  const int ldRow = tid >> 2;          // 0..63
  const int ldCol = (tid & 3) * 4;     // 0,4,8,12
  const int hi    = lane >> 4;         // 0: lanes 0-15, 1: lanes 16-31
  const int mrow  = wm * 16 + (lane & 15);
  const int nIdx  = wn * 32 + (lane & 15);

  for (int k0 = 0; k0 < K; k0 += 16) {
    // coalesced float4 staging of A (guarded) and B (weights, always in range)
    float4 av = make_float4(0.f, 0.f, 0.f, 0.f);
    int ar = mBase + ldRow;
    if (ar < M) av = *(const float4*)&A[(size_t)ar * lda + k0 + ldCol];
    *(float4*)&As[ldRow][ldCol] = av;
    float4 bv = *(const float4*)&B[(size_t)(nBase + ldRow) * K + k0 + ldCol];
    *(float4*)&Bs[ldRow][ldCol] = bv;
    __syncthreads();

#pragma unroll
    for (int kk = 0; kk < 16; kk += 4) {
      // A frag (16x4): VGPR0 = {K+0 | K+2}, VGPR1 = {K+1 | K+3} across lane halves
      v2f a = *(const v2f*)&As[mrow][kk + hi * 2];
      // B frag (4x16): VGPR v, lane l -> K = kk + 2v + (l>>4), N = n0 + (l&15)
      v2f b0 = { Bs[nIdx][kk + hi],      Bs[nIdx][kk + 2 + hi] };
      v2f b1 = { Bs[nIdx + 16][kk + hi], Bs[nIdx + 16][kk + 2 + hi] };
      acc0 = __builtin_amdgcn_wmma_f32_16x16x4_f32(false, a, false, b0,
                                                   (short)0, acc0, false, false);
      acc1 = __builtin_amdgcn_wmma_f32_16x16x4_f32(false, a, false, b1,
                                                   (short)0, acc1, false, false);
    }
    __syncthreads();
  }

  // C/D layout: VGPR r -> M = r (lanes 0-15) / r+8 (lanes 16-31), N = lane&15
  const int colBase = nBase + wn * 32 + (lane & 15);
  const int rOff = hi * 8;
#pragma unroll
  for (int r = 0; r < 8; ++r) {
    int rowi = mBase + wm * 16 + r + rOff;
    if (rowi < M) {
      C[(size_t)rowi * ldc + colBase]      = acc0[r];
      C[(size_t)rowi * ldc + colBase + 16] = acc1[r];
    }
  }
}

// ---------------------------------------------------------------------------
// host-side orchestration
// ---------------------------------------------------------------------------
extern "C" void kernel_launch(void* const* d_in, const int* in_sizes, int n_in,
                              void* d_out, int out_size, void* d_ws, size_t ws_size,
                              hipStream_t stream) {
  const float* x     = (const float*)d_in[0];
  const float* d2an  = (const float*)d_in[1];
  const int*   ei0   = (const int*)d_in[2];
  const float* ea0   = (const float*)d_in[3];
  const int*   ei1   = (const int*)d_in[4];
  const float* ea1   = (const float*)d_in[5];
  const float* Wnode1 = (const float*)d_in[6];
  const float* Wnode2 = (const float*)d_in[7];
  const float* W11 = (const float*)d_in[8];
  const float* W12 = (const float*)d_in[9];
  const float* W21 = (const float*)d_in[10];
  const float* W22 = (const float*)d_in[11];
  const float* W31 = (const float*)d_in[12];
  const float* W32 = (const float*)d_in[13];
  const float* W41 = (const float*)d_in[14];
  const float* W42 = (const float*)d_in[15];
  float* out = (float*)d_out;

  const int* row0 = ei0;          const int* col0 = ei0 + NEDGES;
  const int* row1 = ei1;          const int* col1 = ei1 + NEDGES;

  // workspace carve-up (all offsets multiples of 16 floats -> 16B aligned)
  float* ws = (float*)d_ws;
  size_t off = 0;
  float* dinv0 = ws + off; off += NNODES;
  float* dinv1 = ws + off; off += NNODES;
  float* X  = ws + off; off += (size_t)NNODES * KCAT;    // xcat padded
  float* H  = ws + off; off += (size_t)NNODES * FEMB;    // current layer input
  float* Ag = ws + off; off += (size_t)NNODES * KPAIR;   // a1||a2 aggregation
  float* O0 = ws + off; off += (size_t)NNODES * FEMB;
  float* O1 = ws + off; off += (size_t)NNODES * FEMB;
  float* Wn1p = ws + off; off += 256 * KCAT;
  float* Wn2p = ws + off; off += 256 * KCAT;
  float* Wp0 = ws + off; off += 256 * KPAIR;  // layer1 set0: [W11|W12]
  float* Wp1 = ws + off; off += 256 * KPAIR;  // layer1 set1: [W21|W22]
  float* Wp2 = ws + off; off += 256 * KPAIR;  // layer2 set0: [W31|W32]
  float* Wp3 = ws + off; off += 256 * KPAIR;  // layer2 set1: [W41|W42]
  (void)ws_size; (void)n_in; (void)in_sizes; (void)out_size;

  const int T = 256;
  dim3 gemmGrid((NNODES + 63) / 64, FEMB / 64);

  // --- degrees -> dinv (self loop contributes 1.0) ---
  k_fill<<<(NNODES + T - 1) / T, T, 0, stream>>>(dinv0, NNODES, 1.0f);
  k_fill<<<(NNODES + T - 1) / T, T, 0, stream>>>(dinv1, NNODES, 1.0f);
  k_degcount<<<(NEDGES + T - 1) / T, T, 0, stream>>>(col0, dinv0);
  k_degcount<<<(NEDGES + T - 1) / T, T, 0, stream>>>(col1, dinv1);
  k_rsqrt_inplace<<<(NNODES + T - 1) / T, T, 0, stream>>>(dinv0, NNODES);
  k_rsqrt_inplace<<<(NNODES + T - 1) / T, T, 0, stream>>>(dinv1, NNODES);

  // --- weight packing ---
  long long nxc = (long long)NNODES * KCAT;
  k_xcat<<<(unsigned)((nxc + T - 1) / T), T, 0, stream>>>(x, d2an, X);
  k_padw<<<(256 * KCAT + T - 1) / T, T, 0, stream>>>(Wnode1, Wn1p);
  k_padw<<<(256 * KCAT + T - 1) / T, T, 0, stream>>>(Wnode2, Wn2p);
  k_pair<<<(256 * KPAIR + T - 1) / T, T, 0, stream>>>(W11, W12, Wp0);
  k_pair<<<(256 * KPAIR + T - 1) / T, T, 0, stream>>>(W21, W22, Wp1);
  k_pair<<<(256 * KPAIR + T - 1) / T, T, 0, stream>>>(W31, W32, Wp2);
  k_pair<<<(256 * KPAIR + T - 1) / T, T, 0, stream>>>(W41, W42, Wp3);

  long long nnf = (long long)NNODES * FEMB;
  unsigned gNF = (unsigned)((nnf + T - 1) / T);
  unsigned gSC = (unsigned)(((long long)NEDGES * 64 + T - 1) / T);

  // ================= layer 1 =================
  // edge set 0: h = xcat @ Wnode1^T
  wmma_gemm_nt<<<gemmGrid, T, 0, stream>>>(X, KCAT, Wn1p, H, FEMB, NNODES, KCAT);
  k_inita<<<gNF, T, 0, stream>>>(H, dinv0, Ag);
  k_scatter<<<gSC, T, 0, stream>>>(H, row0, col0, ea0, dinv0, Ag);
  wmma_gemm_nt<<<gemmGrid, T, 0, stream>>>(Ag, KPAIR, Wp0, O0, FEMB, NNODES, KPAIR);

  // edge set 1: h = xcat @ Wnode2^T
  wmma_gemm_nt<<<gemmGrid, T, 0, stream>>>(X, KCAT, Wn2p, H, FEMB, NNODES, KCAT);
  k_inita<<<gNF, T, 0, stream>>>(H, dinv1, Ag);
  k_scatter<<<gSC, T, 0, stream>>>(H, row1, col1, ea1, dinv1, Ag);
  wmma_gemm_nt<<<gemmGrid, T, 0, stream>>>(Ag, KPAIR, Wp1, O1, FEMB, NNODES, KPAIR);

  k_combine<<<gNF, T, 0, stream>>>(O0, O1, H);   // H = layer-2 input

  // ================= layer 2 =================
  k_inita<<<gNF, T, 0, stream>>>(H, dinv0, Ag);
  k_scatter<<<gSC, T, 0, stream>>>(H, row0, col0, ea0, dinv0, Ag);
  wmma_gemm_nt<<<gemmGrid, T, 0, stream>>>(Ag, KPAIR, Wp2, O0, FEMB, NNODES, KPAIR);

  k_inita<<<gNF, T, 0, stream>>>(H, dinv1, Ag);
  k_scatter<<<gSC, T, 0, stream>>>(H, row1, col1, ea1, dinv1, Ag);
  wmma_gemm_nt<<<gemmGrid, T, 0, stream>>>(Ag, KPAIR, Wp3, O1, FEMB, NNODES, KPAIR);

  k_combine<<<gNF, T, 0, stream>>>(O0, O1, out);
}